// PGBM_19670950215706
// MI455X (gfx1250) — compile-verified
//
#include <hip/hip_runtime.h>
#include <hip/hip_bf16.h>

// PGBM split-gain: per-feature 256-bin g/h histograms over 1M samples,
// bin-cumsum, gain formula.
//
// Roofline: 400MB X + 8MB g/h @ 23.3 TB/s => ~17.5us floor. Pass 1 is a
// bandwidth-bound histogram (int4/float4 streaming loads + ds_add_f32 LDS
// atomics + global_atomic_add_f32 flush). Pass 2 (trivial) does the 256-bin
// inclusive scan as a triangular matmul on v_wmma_f32_16x16x4_f32 plus a
// shfl-based column-offset fixup, then the gain formula.

#define NFEAT   100
#define NSAMP   1000000
#define NBINS   256
#define FT      4          // features per histogram block
#define FGRP    (NFEAT / FT)
#define LAMBDA  1.0f

typedef __attribute__((ext_vector_type(2))) float v2f;
typedef __attribute__((ext_vector_type(8))) float v8f;

// ---------------------------------------------------------------- zero ws
__global__ void pgbm_zero_kernel(float* __restrict__ p, int n) {
    int i = blockIdx.x * blockDim.x + threadIdx.x;
    if (i < n) p[i] = 0.0f;
}

// ---------------------------------------------------------------- pass 1
// grid = (chunks, FGRP), block = 256 (8 waves). Each block histograms a
// 4-feature tile over a grid-strided slice of samples into LDS, then
// atomically merges into the global per-feature histograms.
__global__ void __launch_bounds__(256)
pgbm_hist_kernel(const int* __restrict__ X,
                 const float* __restrict__ g,
                 const float* __restrict__ h,
                 float* __restrict__ ghist,
                 float* __restrict__ hhist) {
    __shared__ float lg[FT][NBINS];
    __shared__ float lh[FT][NBINS];

    const int tid = threadIdx.x;
    for (int t = tid; t < FT * NBINS; t += 256) {
        (&lg[0][0])[t] = 0.0f;
        (&lh[0][0])[t] = 0.0f;
    }
    __syncthreads();

    const int f0   = blockIdx.y * FT;
    const int nvec = NSAMP / 4;                       // 250000 int4 groups
    const int step = gridDim.x * 256;

    for (int idx = blockIdx.x * 256 + tid; idx < nvec; idx += step) {
        const float4 gv = ((const float4*)g)[idx];
        const float4 hv = ((const float4*)h)[idx];
#pragma unroll
        for (int f = 0; f < FT; ++f) {
            const int4 xv = ((const int4*)(X + (size_t)(f0 + f) * NSAMP))[idx];
            unsafeAtomicAdd(&lg[f][xv.x], gv.x);
            unsafeAtomicAdd(&lh[f][xv.x], hv.x);
            unsafeAtomicAdd(&lg[f][xv.y], gv.y);
            unsafeAtomicAdd(&lh[f][xv.y], hv.y);
            unsafeAtomicAdd(&lg[f][xv.z], gv.z);
            unsafeAtomicAdd(&lh[f][xv.z], hv.z);
            unsafeAtomicAdd(&lg[f][xv.w], gv.w);
            unsafeAtomicAdd(&lh[f][xv.w], hv.w);
        }
    }
    __syncthreads();

    for (int t = tid; t < FT * NBINS; t += 256) {
        const int f = t >> 8;
        const int b = t & 255;
        unsafeAtomicAdd(&ghist[(size_t)(f0 + f) * NBINS + b], lg[f][b]);
        unsafeAtomicAdd(&hhist[(size_t)(f0 + f) * NBINS + b], lh[f][b]);
    }
}

// ---------------------------------------------------------------- pass 2
// One wave per feature. hist[256] reshaped column-major into V (16x16,
// V[r][c] = hist[c*16+r]). S = L_incl * V gives intra-column inclusive
// scans via 4 chained v_wmma_f32_16x16x4_f32 (K-chunks of 4). Column
// offsets fixed up with a 16-wide shfl scan of the column totals (row 15
// of S). Fragment layouts per CDNA5 ISA 7.12.2:
//   A 16x4 f32:  lane m = lane&15; VGPR0 holds K = 4k + 2*(lane>=16),
//                VGPR1 holds K = 4k + 1 + 2*(lane>=16)
//   B 4x16 f32:  lane n = lane&15; same K striping as A (mirrored)
//   C/D 16x16:   VGPR p: row = p + 8*(lane>=16), col = lane&15
__device__ inline v8f pgbm_wmma_colscan(const float* __restrict__ hist,
                                        int lane) {
    const int m    = lane & 15;            // A row / B col / D col index
    const int koff = (lane >> 4) * 2;      // K striping offset for this half
    v8f acc = {};
#pragma unroll
    for (int k = 0; k < 4; ++k) {
        const int k0 = 4 * k + koff;       // K index carried in VGPR0
        const int k1 = k0 + 1;             // K index carried in VGPR1
        // A = L_incl chunk: A[m][K] = (K <= m)
        v2f a;
        a[0] = (k0 <= m) ? 1.0f : 0.0f;
        a[1] = (k1 <= m) ? 1.0f : 0.0f;
        // B = V chunk: B[K][n] = V[K][n] = hist[n*16 + K]; K0,K1 adjacent
        const float2 bv = *(const float2*)(hist + m * 16 + k0);
        v2f b;
        b[0] = bv.x;
        b[1] = bv.y;
        acc = __builtin_amdgcn_wmma_f32_16x16x4_f32(
            /*neg_a=*/false, a, /*neg_b=*/false, b,
            /*c_mod=*/(short)0, acc, /*reuse_a=*/false, /*reuse_b=*/false);
    }
    return acc;
}

__global__ void __launch_bounds__(32)
pgbm_gain_kernel(const float* __restrict__ ghist,
                 const float* __restrict__ hhist,
                 float* __restrict__ out) {
    const int j    = blockIdx.x;
    const int lane = threadIdx.x;
    const int c    = lane & 15;
    const int half = lane >> 4;

    const float* gh = ghist + (size_t)j * NBINS;
    const float* hh = hhist + (size_t)j * NBINS;

    v8f cg = pgbm_wmma_colscan(gh, lane);   // intra-column inclusive sums (g)
    v8f ch = pgbm_wmma_colscan(hh, lane);   // intra-column inclusive sums (h)

    // Column totals = row 15 of S: VGPR 7, high half, lane 16+c.
    const float gcol = __shfl(cg[7], 16 + c, 32);
    const float hcol = __shfl(ch[7], 16 + c, 32);

    // Inclusive scan of column totals across c (both halves run identically).
    float gs = gcol, hs = hcol;
#pragma unroll
    for (int d = 1; d < 16; d <<= 1) {
        const float gu = __shfl(gs, lane - d, 32);
        const float hu = __shfl(hs, lane - d, 32);
        if (c >= d) { gs += gu; hs += hu; }
    }
    const float G   = __shfl(gs, 15, 32);   // total gradient sum
    const float H   = __shfl(hs, 15, 32);   // total hessian sum
    const float gex = gs - gcol;            // exclusive column offset (g)
    const float hex = hs - hcol;            // exclusive column offset (h)

    const float base = G * G / (H + LAMBDA);

    float res[8];
#pragma unroll
    for (int p = 0; p < 8; ++p) {
        // D[r][c] with r = p + 8*half is inclusive cumsum at bin b = c*16 + r
        const float cumg = cg[p] + gex;
        const float cumh = ch[p] + hex;
        const float Gl = G - cumg;
        const float Hl = H - cumh;
        const float Gr = G - Gl;
        const float Hr = H - Hl;
        res[p] = Gl * Gl / (Hl + LAMBDA) + Gr * Gr / (Hr + LAMBDA) - base;
    }

    // Bins c*16 + 8*half + p, p=0..7 are contiguous -> two float4 stores.
    float* o = out + (size_t)j * NBINS + c * 16 + half * 8;
    *(float4*)(o)     = make_float4(res[0], res[1], res[2], res[3]);
    *(float4*)(o + 4) = make_float4(res[4], res[5], res[6], res[7]);
}

// ---------------------------------------------------------------- launch
extern "C" void kernel_launch(void* const* d_in, const int* in_sizes, int n_in,
                              void* d_out, int out_size, void* d_ws, size_t ws_size,
                              hipStream_t stream) {
    const int*   X = (const int*)d_in[0];     // [100, 1e6] int32 bins
    const float* g = (const float*)d_in[1];   // [1e6] f32
    const float* h = (const float*)d_in[2];   // [1e6] f32
    float* out = (float*)d_out;               // [100, 256] f32

    float* ghist = (float*)d_ws;              // [100, 256]
    float* hhist = ghist + NFEAT * NBINS;     // [100, 256]

    const int nhist = 2 * NFEAT * NBINS;      // 51200 floats
    pgbm_zero_kernel<<<(nhist + 255) / 256, 256, 0, stream>>>(ghist, nhist);

    dim3 hgrid(160, FGRP);                    // 4000 blocks x 8 waves
    pgbm_hist_kernel<<<hgrid, 256, 0, stream>>>(X, g, h, ghist, hhist);

    pgbm_gain_kernel<<<NFEAT, 32, 0, stream>>>(ghist, hhist, out);
}